// PrototypeHead_47845935677591
// MI455X (gfx1250) — compile-verified
//
#include <hip/hip_runtime.h>
#include <hip/hip_bf16.h>

typedef __attribute__((ext_vector_type(2))) float v2f;
typedef __attribute__((ext_vector_type(8))) float v8f;

#define B_N   8192
#define C_N   1000
#define P_N   8
#define D_N   512
#define CP_N  (C_N * P_N)   // 8000

#define MT    128           // M tile (rows of z)
#define NT    64            // N tile (rows of mu = cols of B)
#define KT    64            // K stage depth in LDS
#define LDA   68            // padded LDS stride (floats): stride%64==4 -> conflict-free frag reads
#define LDD   66            // padded stride for epilogue dist tile
#define NTHREADS 128        // 4 wave32; each wave owns a 32x64 sub-tile

// ---------------------------------------------------------------------------
// rowsq: out[r] = sum_d x[r][d]^2 for rows of length 512. One wave32 per row.
// ---------------------------------------------------------------------------
__global__ __launch_bounds__(256) void rowsq_kernel(const float* __restrict__ x,
                                                    float* __restrict__ out,
                                                    int rows) {
    const int lane = threadIdx.x & 31;
    const int row  = blockIdx.x * 8 + (threadIdx.x >> 5);
    if (row >= rows) return;
    const float* r = x + (size_t)row * D_N;
    float s = 0.0f;
    #pragma unroll
    for (int i = 0; i < D_N; i += 128) {          // 4 float4 per lane
        float4 v = *(const float4*)(r + i + lane * 4);
        s += v.x * v.x + v.y * v.y + v.z * v.z + v.w * v.w;
    }
    #pragma unroll
    for (int m = 16; m >= 1; m >>= 1) s += __shfl_xor(s, m, 32);
    if (lane == 0) out[row] = s;
}

// ---------------------------------------------------------------------------
// Main fused kernel: cross = z @ mu^T via V_WMMA_F32_16X16X4_F32,
// epilogue: min_p (z2 - 2*cross + mu2), write logits=-min and min.
// Grid: (125 N-tiles, 64 M-tiles), 128 threads = 4 wave32,
// each wave computes a 32x64 sub-tile (8 accumulators -> 8 WMMAs / 6 LDS loads
// per K-step).
// ---------------------------------------------------------------------------
__global__ __launch_bounds__(NTHREADS) void proto_head_kernel(
    const float* __restrict__ z,    // [B_N, D_N]
    const float* __restrict__ mu,   // [CP_N, D_N]
    const float* __restrict__ z2,   // [B_N]
    const float* __restrict__ mu2,  // [CP_N]
    float* __restrict__ out)        // [2, B_N, C_N]
{
    __shared__ float smem[MT * LDA + NT * LDA];   // 13056 floats = 52.2 KB
    float* zs = smem;                             // z tile  [MT][LDA]
    float* ms = smem + MT * LDA;                  // mu tile [NT][LDA]

    const int t     = threadIdx.x;
    const int wave  = t >> 5;        // 0..3
    const int lane  = t & 31;
    const int lo    = lane & 15;     // M (or N) index within 16
    const int hi    = lane >> 4;     // selects K pair 2*hi..2*hi+1
    const int bn    = blockIdx.x;    // 0..124
    const int bm    = blockIdx.y;    // 0..63
    const int waveM = wave * 32;     // 0,32,64,96 (each wave: 32 rows x all 64 cols)

    const float* zg = z  + (size_t)(bm * MT) * D_N;
    const float* mg = mu + (size_t)(bn * NT) * D_N;

    v8f acc[2][4] = {};

    for (int ks = 0; ks < D_N; ks += KT) {
        // Stage z tile: 128x64 f32 = 2048 float4, 16 per thread (coalesced)
        #pragma unroll
        for (int p = 0; p < 16; ++p) {
            int idx = p * NTHREADS + t;
            int row = idx >> 4;            // 16 float4 per row
            int c4  = (idx & 15) << 2;
            float4 v = *(const float4*)(zg + (size_t)row * D_N + ks + c4);
            *(float4*)(zs + row * LDA + c4) = v;   // (row*68+c4)*4 is 16B aligned
        }
        // Stage mu tile: 64x64 f32 = 1024 float4, 8 per thread
        #pragma unroll
        for (int p = 0; p < 8; ++p) {
            int idx = p * NTHREADS + t;
            int row = idx >> 4;
            int c4  = (idx & 15) << 2;
            float4 v = *(const float4*)(mg + (size_t)row * D_N + ks + c4);
            *(float4*)(ms + row * LDA + c4) = v;
        }
        __syncthreads();

        // 16 K-steps of 4; 8 independent WMMAs per step (32x64 per wave)
        #pragma unroll
        for (int kk = 0; kk < KT; kk += 4) {
            // A frags: lane holds z[waveM + i*16 + lo][kk + 2*hi + {0,1}]
            v2f a0 = *(const v2f*)(zs + (waveM +      lo) * LDA + kk + 2 * hi);
            v2f a1 = *(const v2f*)(zs + (waveM + 16 + lo) * LDA + kk + 2 * hi);
            // B frags: lane holds B[kk+2*hi+{0,1}][j*16 + lo] = mu[n][k]
            v2f b0 = *(const v2f*)(ms + (      lo) * LDA + kk + 2 * hi);
            v2f b1 = *(const v2f*)(ms + (16 + lo) * LDA + kk + 2 * hi);
            v2f b2 = *(const v2f*)(ms + (32 + lo) * LDA + kk + 2 * hi);
            v2f b3 = *(const v2f*)(ms + (48 + lo) * LDA + kk + 2 * hi);

            acc[0][0] = __builtin_amdgcn_wmma_f32_16x16x4_f32(
                false, a0, false, b0, (short)0, acc[0][0], false, false);
            acc[0][1] = __builtin_amdgcn_wmma_f32_16x16x4_f32(
                false, a0, false, b1, (short)0, acc[0][1], false, false);
            acc[0][2] = __builtin_amdgcn_wmma_f32_16x16x4_f32(
                false, a0, false, b2, (short)0, acc[0][2], false, false);
            acc[0][3] = __builtin_amdgcn_wmma_f32_16x16x4_f32(
                false, a0, false, b3, (short)0, acc[0][3], false, false);
            acc[1][0] = __builtin_amdgcn_wmma_f32_16x16x4_f32(
                false, a1, false, b0, (short)0, acc[1][0], false, false);
            acc[1][1] = __builtin_amdgcn_wmma_f32_16x16x4_f32(
                false, a1, false, b1, (short)0, acc[1][1], false, false);
            acc[1][2] = __builtin_amdgcn_wmma_f32_16x16x4_f32(
                false, a1, false, b2, (short)0, acc[1][2], false, false);
            acc[1][3] = __builtin_amdgcn_wmma_f32_16x16x4_f32(
                false, a1, false, b3, (short)0, acc[1][3], false, false);
        }
        __syncthreads();
    }

    // Spill accumulators to LDS (reuse staging memory; all reads done).
    // C/D layout: lanes 0-15 -> M=r, N=lane; lanes 16-31 -> M=8+r, N=lane-16.
    float* dsT = smem;   // dist tile [MT][LDD]
    #pragma unroll
    for (int msb = 0; msb < 2; ++msb)
        #pragma unroll
        for (int nsb = 0; nsb < 4; ++nsb)
            #pragma unroll
            for (int r = 0; r < 8; ++r) {
                int row = waveM + msb * 16 + 8 * hi + r;
                int col = nsb * 16 + lo;
                dsT[row * LDD + col] = acc[msb][nsb][r];
            }
    __syncthreads();

    // Epilogue: 128 rows x 8 classes per block = 1024 tasks, 8 per thread.
    const int baseB = bm * MT;
    const int baseC = bn * (NT / P_N);       // bn*8
    const size_t outStride = (size_t)B_N * C_N;
    #pragma unroll
    for (int q = t; q < MT * (NT / P_N); q += NTHREADS) {
        int m  = q >> 3;
        int cl = q & 7;
        float zz   = z2[baseB + m];
        float best = 3.402823466e38f;
        #pragma unroll
        for (int j = 0; j < P_N; ++j) {
            float cr = dsT[m * LDD + cl * P_N + j];
            float d  = zz - 2.0f * cr + mu2[bn * NT + cl * P_N + j];
            best = fminf(best, d);
        }
        size_t o = (size_t)(baseB + m) * C_N + (baseC + cl);
        out[o]             = -best;   // logits
        out[outStride + o] =  best;   // min_dist_sq
    }
}

// ---------------------------------------------------------------------------
extern "C" void kernel_launch(void* const* d_in, const int* in_sizes, int n_in,
                              void* d_out, int out_size, void* d_ws, size_t ws_size,
                              hipStream_t stream) {
    (void)in_sizes; (void)n_in; (void)out_size; (void)ws_size;
    const float* z  = (const float*)d_in[0];   // [8192, 512]
    const float* mu = (const float*)d_in[1];   // [1000, 8, 512]
    float* out = (float*)d_out;                // [2, 8192, 1000]
    float* z2  = (float*)d_ws;                 // [8192]
    float* mu2 = z2 + B_N;                     // [8000]

    rowsq_kernel<<<B_N / 8,  256, 0, stream>>>(z,  z2,  B_N);
    rowsq_kernel<<<CP_N / 8, 256, 0, stream>>>(mu, mu2, CP_N);

    dim3 grid(CP_N / NT, B_N / MT);            // (125, 64)
    proto_head_kernel<<<grid, NTHREADS, 0, stream>>>(z, mu, z2, mu2, out);
}